// Net_27882927686182
// MI455X (gfx1250) — compile-verified
//
#include <hip/hip_runtime.h>

#define M_NODES 4096
#define N_LAYERS 32
#define K_DEG 64

#define TPB 128             // 4 waves of 32 (wave32)
#define NODES_PER_BLOCK 64  // 4 waves x 16-node WMMA tiles

typedef __attribute__((ext_vector_type(2))) float v2f;
typedef __attribute__((ext_vector_type(8))) float v8f;
typedef __attribute__((ext_vector_type(4))) int   v4i_t;

#ifdef __HIP_DEVICE_COMPILE__
#if __has_builtin(__builtin_amdgcn_global_load_async_to_lds_b128)
#define HAVE_ASYNC_LDS 1
#endif
#endif

// ---------------- input layer: v0[m] = relu(w_in[m]*x[m] + b_in[m]) ----------------
__global__ __launch_bounds__(256) void input_kernel(const float* __restrict__ x,
                                                    const float* __restrict__ w_in,
                                                    const float* __restrict__ b_in,
                                                    float* __restrict__ v0) {
    int m = blockIdx.x * blockDim.x + threadIdx.x;
    if (m < M_NODES) {
        float t = fmaf(w_in[m], x[m], b_in[m]);
        v0[m] = t > 0.0f ? t : 0.0f;
    }
}

// ---------------- internal layer via V_WMMA_F32_16X16X4_F32 ----------------
// Each wave computes a 16-node tile: A[m][k] = w[m][k] * v[idx[m][k]] in the
// ISA 16x4 f32 A layout; B = ones(4x16) so D[m][n] = sum_k A[m][k]. Two
// independent 8-deep WMMA chains cover K=64 with f32 accumulation.
__global__ __launch_bounds__(TPB) void layer_kernel(
        const float* __restrict__ w,        // [M][K], pre-offset to this layer
        const float* __restrict__ b,        // [M]
        const int*   __restrict__ idx,      // [M][K]
        const float* __restrict__ v_in,     // [M]
        float*       __restrict__ v_out,    // [M]
        const float* __restrict__ w_next,   // next layer's weights (prefetch)
        const int*   __restrict__ idx_next, // next layer's indices (prefetch)
        int has_next) {
    __shared__ __align__(16) float vlds[M_NODES];   // 16 KB of 320 KB/WGP

    const int tid = threadIdx.x;

    // Stage full previous-layer value vector into LDS.
#ifdef HAVE_ASYNC_LDS
    {
        typedef __attribute__((address_space(1))) v4i_t g_v4i;
        typedef __attribute__((address_space(3))) v4i_t l_v4i;
        #pragma unroll
        for (int j = 0; j < M_NODES / (TPB * 4); ++j) {
            const int i = j * (TPB * 4) + tid * 4;     // 16B per lane, coalesced
            __builtin_amdgcn_global_load_async_to_lds_b128(
                (g_v4i*)(uintptr_t)(v_in + i),
                (l_v4i*)(uint32_t)(uintptr_t)(&vlds[i]),
                0, 0);
        }
    }
#else
    {
        #pragma unroll
        for (int j = 0; j < M_NODES / (TPB * 4); ++j) {
            const int i = j * (TPB * 4) + tid * 4;
            *(float4*)(&vlds[i]) = *(const float4*)(v_in + i);
        }
    }
#endif

    // Hide the serialized layer chain: prefetch next layer's slice for this
    // block (128B cache line per thread covers the 16KB w slice + 16KB idx slice).
    if (has_next) {
        size_t base = (size_t)blockIdx.x * NODES_PER_BLOCK * K_DEG;
        __builtin_prefetch(w_next + base + (size_t)tid * 32, 0, 3);
        __builtin_prefetch(idx_next + base + (size_t)tid * 32, 0, 3);
    }

#ifdef HAVE_ASYNC_LDS
#if __has_builtin(__builtin_amdgcn_s_wait_asynccnt)
    __builtin_amdgcn_s_wait_asynccnt(0);
#else
    asm volatile("s_wait_asynccnt 0x0" ::: "memory");
#endif
#endif
    __syncthreads();

    const int wave = tid >> 5;        // 0..3
    const int lane = tid & 31;
    const int half = lane >> 4;       // 0: k pair {4i,4i+1}; 1: {4i+2,4i+3}
    const int lr   = lane & 15;       // row within 16-node tile
    const int tileBase = blockIdx.x * NODES_PER_BLOCK + wave * 16;
    const int m = tileBase + lr;

    const float* wrow = w   + (size_t)m * K_DEG;
    const int*   irow = idx + (size_t)m * K_DEG;

    v8f acc0 = {0.f, 0.f, 0.f, 0.f, 0.f, 0.f, 0.f, 0.f};
    v8f acc1 = {0.f, 0.f, 0.f, 0.f, 0.f, 0.f, 0.f, 0.f};
    v2f ones;
    ones.x = 1.0f;
    ones.y = 1.0f;

    // Two independent accumulation chains to hide WMMA RAW latency.
    #pragma unroll
    for (int i = 0; i < 16; i += 2) {
        const int ka = 4 * i + 2 * half;          // 8B-aligned pair -> b64 loads
        const int kb = ka + 4;
        float wa0 = wrow[ka], wa1 = wrow[ka + 1];
        float wb0 = wrow[kb], wb1 = wrow[kb + 1];
        int ia0 = irow[ka], ia1 = irow[ka + 1];
        int ib0 = irow[kb], ib1 = irow[kb + 1];
        v2f a, c;
        a.x = wa0 * vlds[ia0];                    // ds_load_b32 gathers
        a.y = wa1 * vlds[ia1];
        c.x = wb0 * vlds[ib0];
        c.y = wb1 * vlds[ib1];
        acc0 = __builtin_amdgcn_wmma_f32_16x16x4_f32(
                false, a, false, ones, (short)0, acc0, false, false);
        acc1 = __builtin_amdgcn_wmma_f32_16x16x4_f32(
                false, c, false, ones, (short)0, acc1, false, false);
    }
    v8f acc = acc0 + acc1;

    // D layout: VGPR r -> row r (lanes 0-15), row r+8 (lanes 16-31); every
    // column holds the full row sum, so lanes 0 and 16 own all 16 rows.
    if (lr == 0) {
        const int rowBase = tileBase + half * 8;
        #pragma unroll
        for (int r = 0; r < 8; ++r) {
            float s = acc[r] + b[rowBase + r];
            v_out[rowBase + r] = 1.0f / (1.0f + __expf(-s));
        }
    }
}

// ---------------- final: out = v[M-1] ----------------
__global__ void output_kernel(const float* __restrict__ v, float* __restrict__ out) {
    out[0] = v[M_NODES - 1];
}

extern "C" void kernel_launch(void* const* d_in, const int* in_sizes, int n_in,
                              void* d_out, int out_size, void* d_ws, size_t ws_size,
                              hipStream_t stream) {
    const float* x     = (const float*)d_in[0];   // [M,1]
    const float* w_in  = (const float*)d_in[1];   // [M,1]
    const float* b_in  = (const float*)d_in[2];   // [M]
    const float* w     = (const float*)d_in[3];   // [L,M,K]
    const float* b     = (const float*)d_in[4];   // [L,M]
    const int*   igraf = (const int*)d_in[5];     // [L,M,K]

    float* v0 = (float*)d_ws;                     // ping-pong value vectors
    float* v1 = v0 + M_NODES;                     // 32 KB total scratch

    input_kernel<<<M_NODES / 256, 256, 0, stream>>>(x, w_in, b_in, v0);

    for (int l = 0; l < N_LAYERS; ++l) {
        const float* vin  = (l & 1) ? v1 : v0;
        float*       vout = (l & 1) ? v0 : v1;
        const float* wl = w     + (size_t)l * M_NODES * K_DEG;
        const int*   il = igraf + (size_t)l * M_NODES * K_DEG;
        const float* bl = b     + (size_t)l * M_NODES;
        const int has_next = (l + 1 < N_LAYERS) ? 1 : 0;
        const float* wn = wl + (has_next ? (size_t)M_NODES * K_DEG : 0);
        const int*   in_ = il + (has_next ? (size_t)M_NODES * K_DEG : 0);
        layer_kernel<<<M_NODES / NODES_PER_BLOCK, TPB, 0, stream>>>(
            wl, bl, il, vin, vout, wn, in_, has_next);
    }

    // Layer 31 (odd) wrote into v0.
    output_kernel<<<1, 1, 0, stream>>>(v0, (float*)d_out);
}